// MedianAggregator_23201413333259
// MI455X (gfx1250) — compile-verified
//
#include <hip/hip_runtime.h>

typedef float v2f __attribute__((ext_vector_type(2)));
typedef float v8f __attribute__((ext_vector_type(8)));

#define TILE_M     16
#define FEAT       128
#define NEIGH_S    32
#define UNITS      128
#define OUT_W      256
#define LDS_STRIDE 132   // 128 + 4-float pad: 16-row b64 column reads hit distinct banks

// ---- Batcher merge-exchange sorting network for 16 inputs: 63 comparators ----
// (Knuth TAOCP 5.2.2 Algorithm M, generated at compile time so every index is
//  a constant after unrolling -> the 32 values live entirely in VGPRs.)
struct Net16 {
    unsigned char a[63];
    unsigned char b[63];
    int count;
};

constexpr Net16 make_net16() {
    Net16 n{};
    int c = 0;
    for (int p = 8; p > 0; p >>= 1) {
        int q = 8, r = 0, d = p;
        while (true) {
            for (int i = 0; i + d < 16; ++i)
                if ((i & p) == r) { n.a[c] = (unsigned char)i;
                                    n.b[c] = (unsigned char)(i + d); ++c; }
            if (q == p) break;
            d = q - p; q >>= 1; r = p;
        }
    }
    n.count = c;   // 63
    return n;
}

__device__ __forceinline__ void cas(float& x, float& y) {
    float lo = fminf(x, y);
    float hi = fmaxf(x, y);
    x = lo; y = hi;
}

__device__ __forceinline__ void sort16(float* v) {
    constexpr Net16 net = make_net16();
    static_assert(make_net16().count == 63, "Batcher-16 must have 63 comparators");
#pragma unroll
    for (int c = 0; c < 63; ++c)
        cas(v[net.a[c]], v[net.b[c]]);
}

// rank-16 (0-indexed, 17th smallest) of v[0..31]: sort both halves ascending,
// then merged[16] = max over i+j==16 of min(A[i], B[j])  (out-of-range -> skip).
__device__ __forceinline__ float rank16_of_32(float* v) {
    sort16(v);
    sort16(v + 16);
    float m = fmaxf(v[0], v[16]);          // i=0 (B[16]=+inf) and i=16 (A[16]=+inf)
#pragma unroll
    for (int i = 1; i <= 15; ++i)
        m = fmaxf(m, fminf(v[i], v[16 + 16 - i]));
    return m;
}

__global__ __launch_bounds__(256)
void median_sage_fused(const float* __restrict__ x,
                       const float* __restrict__ neigh,
                       const float* __restrict__ wself,
                       const float* __restrict__ wneigh,
                       const float* __restrict__ bias,
                       float* __restrict__ out,
                       int nrows)
{
    __shared__ float xT[TILE_M * LDS_STRIDE];
    __shared__ float mT[TILE_M * LDS_STRIDE];

    const int  t  = threadIdx.x;
    const long n0 = (long)blockIdx.x * TILE_M;

    // ---------------- stage x tile into LDS (coalesced) ----------------
#pragma unroll
    for (int i = 0; i < 8; ++i) {
        int idx = t + i * 256;                 // 0..2047
        int r = idx >> 7, f = idx & 127;
        long row = n0 + r; if (row >= nrows) row = nrows - 1;
        xT[r * LDS_STRIDE + f] = x[row * FEAT + f];
    }

    // -------- median phase: 8 (row,f) columns per thread, streamed NT --------
    // neigh_x is 819 MB read exactly once: non-temporal keeps the 192 MB L2
    // free for the weight matrices (128 KB, shared by all 3125 blocks) and x.
    {
        const int f     = t & 127;             // lanes consecutive in f -> coalesced
        const int rbase = (t >> 7) * 8;        // rows 0-7 or 8-15
#pragma unroll
        for (int rr = 0; rr < 8; ++rr) {
            const int r = rbase + rr;
            long row = n0 + r; if (row >= nrows) row = nrows - 1;
            const float* col = neigh + (row * NEIGH_S) * FEAT + f;
            float v[32];
#pragma unroll
            for (int s = 0; s < NEIGH_S; ++s)  // 32 independent NT loads in flight
                v[s] = __builtin_nontemporal_load(col + (long)s * FEAT);
            mT[r * LDS_STRIDE + f] = rank16_of_32(v);
        }
    }
    __syncthreads();

    // -------- WMMA phase: wave w owns output cols [16w,16w+16) of each half --------
    const int lane  = t & 31;
    const int wv    = t >> 5;                  // 0..7
    const int col0  = wv * 16;
    const int mrow  = lane & 15;               // M (for A) / N (for B)
    const int khalf = (lane >> 4) * 2;         // lanes 16-31 hold K+2,K+3

    const float* aS = xT + mrow * LDS_STRIDE + khalf;
    const float* aM = mT + mrow * LDS_STRIDE + khalf;
    const float* bS = wself  + col0 + mrow;    // + k*UNITS
    const float* bN = wneigh + col0 + mrow;

    v8f cs = {};
    v8f cn = {};
#pragma unroll
    for (int k = 0; k < FEAT; k += 4) {
        v2f a_s = *(const v2f*)(aS + k);       // ds_load_(2addr_)b64, conflict-free
        v2f a_m = *(const v2f*)(aM + k);
        const int kk = k + khalf;
        v2f b_s; b_s.x = bS[(long)kk * UNITS]; b_s.y = bS[(long)(kk + 1) * UNITS];
        v2f b_n; b_n.x = bN[(long)kk * UNITS]; b_n.y = bN[(long)(kk + 1) * UNITS];
        cs = __builtin_amdgcn_wmma_f32_16x16x4_f32(false, a_s, false, b_s,
                                                   (short)0, cs, false, false);
        cn = __builtin_amdgcn_wmma_f32_16x16x4_f32(false, a_m, false, b_n,
                                                   (short)0, cn, false, false);
    }

    // ---------------- epilogue: bias + ReLU + NT store ----------------
    const int ncol  = lane & 15;
    const int mbase = (lane >> 4) * 8;         // C layout: lanes 16-31 hold M = j+8
    const float bsv = bias[col0 + ncol];
    const float bnv = bias[UNITS + col0 + ncol];
#pragma unroll
    for (int j = 0; j < 8; ++j) {
        long row = n0 + mbase + j;
        if (row < nrows) {
            __builtin_nontemporal_store(fmaxf(cs[j] + bsv, 0.0f),
                                        out + row * OUT_W + col0 + ncol);
            __builtin_nontemporal_store(fmaxf(cn[j] + bnv, 0.0f),
                                        out + row * OUT_W + UNITS + col0 + ncol);
        }
    }
}

extern "C" void kernel_launch(void* const* d_in, const int* in_sizes, int n_in,
                              void* d_out, int out_size, void* d_ws, size_t ws_size,
                              hipStream_t stream) {
    const float* x      = (const float*)d_in[0];
    const float* neigh  = (const float*)d_in[1];
    const float* wself  = (const float*)d_in[2];
    const float* wneigh = (const float*)d_in[3];
    const float* bias   = (const float*)d_in[4];
    float* out          = (float*)d_out;

    const int nrows  = in_sizes[0] / FEAT;               // 50000
    const int blocks = (nrows + TILE_M - 1) / TILE_M;    // 3125

    median_sage_fused<<<blocks, 256, 0, stream>>>(x, neigh, wself, wneigh, bias,
                                                  out, nrows);
}